// QuantumRNNLayer_10943576670881
// MI455X (gfx1250) — compile-verified
//
#include <hip/hip_runtime.h>
#include <hip/hip_bf16.h>
#include <math.h>

#define NQ 4
#define NL 2
#define EMB 512
#define BATCH 512
#define TSTEPS 128
#define PI_F 3.14159265358979f

typedef float v2f __attribute__((ext_vector_type(2)));
typedef float v8f __attribute__((ext_vector_type(8)));

// ---------------------------------------------------------------------------
// Kernel 1: pre[r][q] = sum_e x[r][e] * W_enc[e][q] + b_enc[q],  r = b*T + t
// One wave per 16-row tile; K-loop of V_WMMA_F32_16X16X4_F32 (fp32-exact).
// N padded 4 -> 16 with zero columns (memory-bound phase, waste is free).
// ---------------------------------------------------------------------------
__global__ __launch_bounds__(256) void qrnn_encode_gemm(
    const float* __restrict__ x, const float* __restrict__ W_enc,
    const float* __restrict__ b_enc, float* __restrict__ pre)
{
  const int lane = threadIdx.x & 31;
  const int wave = threadIdx.x >> 5;
  const int baseRow = blockIdx.x * 128 + wave * 16;
  const int m   = lane & 15;     // A row / D column index for this lane
  const int hi  = lane >> 4;     // half-wave select (K pair 0/1 vs 2/3)
  const int hi2 = hi << 1;

  const float* xrow = x + (size_t)(baseRow + m) * EMB;
  const int nCol = m;                       // B/D column
  const int nClamp = (nCol < 4) ? nCol : 3; // in-bounds, no divergence

  v8f acc = {};
  #pragma unroll 4
  for (int k = 0; k < EMB; k += 4) {
    v2f a;
    a.x = xrow[k + hi2];
    a.y = xrow[k + hi2 + 1];
    float w0 = W_enc[(k + hi2)     * NQ + nClamp];
    float w1 = W_enc[(k + hi2 + 1) * NQ + nClamp];
    v2f b;
    b.x = (nCol < 4) ? w0 : 0.0f;
    b.y = (nCol < 4) ? w1 : 0.0f;
    acc = __builtin_amdgcn_wmma_f32_16x16x4_f32(false, a, false, b, (short)0,
                                                acc, false, false);
  }

  if (nCol < 4) {
    const float be = b_enc[nCol];
    #pragma unroll
    for (int v = 0; v < 8; ++v) {
      const int row = baseRow + hi * 8 + v;   // D: VGPR v = M (v + 8*hi)
      pre[row * NQ + nCol] = acc[v] + be;
    }
  }
}

// ---------------------------------------------------------------------------
// Fast device math: avoid ocml full-range argument reduction (SALU-heavy).
// ---------------------------------------------------------------------------
__device__ __forceinline__ float fast_tanh(float z) {
  // tanh(z) = sign(z) * (1 - 2 / (exp(2|z|) + 1)); exp->inf gives r->1. OK.
  const float az = __builtin_fabsf(z);
  const float e  = __expf(2.0f * az);               // v_exp_f32 path
  const float r  = 1.0f - 2.0f * __builtin_amdgcn_rcpf(e + 1.0f);
  return __builtin_copysignf(r, z);
}

// ds_swizzle_b32 helpers (single DS op, immediate pattern, no index VALU):
//   group-of-32 mode: src_lane = ((lane & AND) | OR) ^ XOR
#define SWZ(v, imm) \
  __int_as_float(__builtin_amdgcn_ds_swizzle(__float_as_int(v), (imm)))
#define IMM_XOR(m)   (((m) << 10) | 0x1f)          // pair exchange lane^m
#define IMM_BCAST(q) (((q) << 5)  | 0x10)          // broadcast lane (base16|q)

// ---------------------------------------------------------------------------
// Kernel 2: sequential scan. 16 lanes = one batch element's 4-qubit
// statevector (amplitude index a; qubit q <-> bit (3-q) of a). Gates via
// ds_swizzle pair exchange; <Z_q> via one 4-stage Walsh-Hadamard butterfly.
// ---------------------------------------------------------------------------
struct Cplx { float re, im; };
__device__ __forceinline__ Cplx cmul(Cplx a, Cplx b) {
  return {a.re * b.re - a.im * b.im, a.re * b.im + a.im * b.re};
}

__global__ __launch_bounds__(256) void qrnn_recurrent(
    const float* __restrict__ pre, const float* __restrict__ W_enc,
    const float* __restrict__ theta, const float* __restrict__ W_read,
    const float* __restrict__ b_read, float* __restrict__ out)
{
  const int tid  = blockIdx.x * blockDim.x + threadIdx.x;
  const int elem = tid >> 4;   // batch element
  const int a    = tid & 15;   // amplitude index of this lane

  // ---- fold constant per-(layer,qubit) gate U = RY*RZ*RX into two per-lane
  //      complex coefficients: new = ca*own + cb*partner ----
  Cplx uca[NL * NQ], ucb[NL * NQ];
  #pragma unroll
  for (int l = 0; l < NL; ++l) {
    #pragma unroll
    for (int q = 0; q < NQ; ++q) {
      const int g = l * NQ + q;
      const float t0 = 0.5f * theta[g * 3 + 0];
      const float t1 = 0.5f * theta[g * 3 + 1];
      const float t2 = 0.5f * theta[g * 3 + 2];
      const float c0 = __cosf(t0), s0 = __sinf(t0);
      const float c1 = __cosf(t1), s1 = __sinf(t1);
      const float c2 = __cosf(t2), s2 = __sinf(t2);
      const Cplx em = {c1, -s1}, ep = {c1, s1};
      const Cplx rx00 = {c0, 0.f}, rx01 = {0.f, -s0};
      const Cplx a00 = cmul(em, rx00), a01 = cmul(em, rx01);
      const Cplx a10 = cmul(ep, rx01), a11 = cmul(ep, rx00);
      const Cplx u00 = {c2 * a00.re - s2 * a10.re, c2 * a00.im - s2 * a10.im};
      const Cplx u01 = {c2 * a01.re - s2 * a11.re, c2 * a01.im - s2 * a11.im};
      const Cplx u10 = {s2 * a00.re + c2 * a10.re, s2 * a00.im + c2 * a10.im};
      const Cplx u11 = {s2 * a01.re + c2 * a11.re, s2 * a01.im + c2 * a11.im};
      const int b = (a >> (3 - q)) & 1;
      uca[g] = b ? u11 : u00;
      ucb[g] = b ? u10 : u01;
    }
  }

  // ---- small constant weights into registers ----
  float Wh[4][4], Wr[4][4], br[4];
  #pragma unroll
  for (int j = 0; j < 4; ++j) {
    #pragma unroll
    for (int q = 0; q < 4; ++q) {
      Wh[j][q] = W_enc[(EMB + j) * NQ + q];
      Wr[j][q] = W_read[j * NQ + q];
    }
    br[j] = b_read[j];
  }

  const float* preB = pre + (size_t)elem * TSTEPS * NQ;
  const int qSelf = a & 3;   // which angle this lane computes
  float h0 = 0.f, h1 = 0.f, h2 = 0.f, h3 = 0.f;
  float pv = preB[qSelf];    // prefetched pre for t=0

  for (int t = 0; t < TSTEPS; ++t) {
    const float pvCur = pv;
    const int tn = (t + 1 < TSTEPS) ? (t + 1) : t;
    pv = preB[tn * NQ + qSelf];  // prefetch next step (L2-resident, 1 MB)

    const float z = pvCur + h0 * Wh[0][qSelf] + h1 * Wh[1][qSelf] +
                    h2 * Wh[2][qSelf] + h3 * Wh[3][qSelf];
    const float ang = fast_tanh(z) * PI_F;
    const float half = 0.5f * ang;               // in [-pi/2, pi/2]
    const float ce = __cosf(half);
    const float se = __sinf(half);

    float sre = (a == 0) ? 1.0f : 0.0f;          // |0000>
    float sim = 0.0f;

    // ---- encoder RY(ang[q]) on wire q (mask = 1<<(3-q)) ----
    #define ENC_GATE(Q, MSK)                                         \
      do {                                                           \
        const float c  = SWZ(ce, IMM_BCAST(Q));                      \
        const float s  = SWZ(se, IMM_BCAST(Q));                      \
        const float pr = SWZ(sre, IMM_XOR(MSK));                     \
        const float pi = SWZ(sim, IMM_XOR(MSK));                     \
        const float sb = (a & (MSK)) ? s : -s;                       \
        sre = c * sre + sb * pr;                                     \
        sim = c * sim + sb * pi;                                     \
      } while (0)
    ENC_GATE(0, 8); ENC_GATE(1, 4); ENC_GATE(2, 2); ENC_GATE(3, 1);
    #undef ENC_GATE

    // ---- variational layers: U(theta) gates then CNOT chain ----
    #define UTH_GATE(G, MSK)                                         \
      do {                                                           \
        const float pr = SWZ(sre, IMM_XOR(MSK));                     \
        const float pi = SWZ(sim, IMM_XOR(MSK));                     \
        const float nr = uca[G].re * sre - uca[G].im * sim +         \
                         ucb[G].re * pr  - ucb[G].im * pi;           \
        const float ni = uca[G].re * sim + uca[G].im * sre +         \
                         ucb[G].re * pi  + ucb[G].im * pr;           \
        sre = nr; sim = ni;                                          \
      } while (0)
    #define CNOT(MC, MT)                                             \
      do {                                                           \
        const float pr = SWZ(sre, IMM_XOR(MT));                      \
        const float pi = SWZ(sim, IMM_XOR(MT));                      \
        const bool ct = (a & (MC)) != 0;                             \
        sre = ct ? pr : sre;                                         \
        sim = ct ? pi : sim;                                         \
      } while (0)
    UTH_GATE(0, 8); UTH_GATE(1, 4); UTH_GATE(2, 2); UTH_GATE(3, 1);
    CNOT(8, 4);     CNOT(4, 2);     CNOT(2, 1);
    UTH_GATE(4, 8); UTH_GATE(5, 4); UTH_GATE(6, 2); UTH_GATE(7, 1);
    CNOT(8, 4);     CNOT(4, 2);     CNOT(2, 1);
    #undef UTH_GATE
    #undef CNOT

    // ---- probabilities -> all four <Z_q> via one 16-point WHT ----
    float p = sre * sre + sim * sim;
    #define WHT_STAGE(M)                                             \
      do {                                                           \
        const float pp = SWZ(p, IMM_XOR(M));                         \
        p = (a & (M)) ? (pp - p) : (p + pp);                         \
      } while (0)
    WHT_STAGE(1); WHT_STAGE(2); WHT_STAGE(4); WHT_STAGE(8);
    #undef WHT_STAGE

    // ev[q] = WHT coefficient at amplitude index 1<<(3-q)
    const float ev0 = SWZ(p, IMM_BCAST(8));
    const float ev1 = SWZ(p, IMM_BCAST(4));
    const float ev2 = SWZ(p, IMM_BCAST(2));
    const float ev3 = SWZ(p, IMM_BCAST(1));

    // ---- readout: h = tanh(ev @ W_read + b_read) (replicated per lane) ----
    h0 = fast_tanh(ev0*Wr[0][0] + ev1*Wr[1][0] + ev2*Wr[2][0] + ev3*Wr[3][0] + br[0]);
    h1 = fast_tanh(ev0*Wr[0][1] + ev1*Wr[1][1] + ev2*Wr[2][1] + ev3*Wr[3][1] + br[1]);
    h2 = fast_tanh(ev0*Wr[0][2] + ev1*Wr[1][2] + ev2*Wr[2][2] + ev3*Wr[3][2] + br[2]);
    h3 = fast_tanh(ev0*Wr[0][3] + ev1*Wr[1][3] + ev2*Wr[2][3] + ev3*Wr[3][3] + br[3]);
  }

  if (a < 4) {
    const float hv = (a == 0) ? h0 : (a == 1) ? h1 : (a == 2) ? h2 : h3;
    out[elem * NQ + a] = hv;
  }
}

// ---------------------------------------------------------------------------
extern "C" void kernel_launch(void* const* d_in, const int* in_sizes, int n_in,
                              void* d_out, int out_size, void* d_ws, size_t ws_size,
                              hipStream_t stream) {
  const float* x      = (const float*)d_in[0];   // (B, T, E)
  const float* W_enc  = (const float*)d_in[1];   // (E+4, 4)
  const float* b_enc  = (const float*)d_in[2];   // (4,)
  const float* theta  = (const float*)d_in[3];   // (L, 4, 3)
  const float* W_read = (const float*)d_in[4];   // (4, 4)
  const float* b_read = (const float*)d_in[5];   // (4,)
  float* out = (float*)d_out;                    // (B, 4)
  float* pre = (float*)d_ws;                     // (B*T, 4) = 1 MB scratch

  // Phase 1: hoisted encoder GEMM (memory-bound, WMMA f32 16x16x4).
  qrnn_encode_gemm<<<(BATCH * TSTEPS) / 128, 256, 0, stream>>>(x, W_enc, b_enc, pre);

  // Phase 2: sequential scan, 16 lanes per batch element -> 8192 threads.
  qrnn_recurrent<<<(BATCH * 16) / 256, 256, 0, stream>>>(pre, W_enc, theta,
                                                         W_read, b_read, out);
}